// StructuredSparsityMask_75943611727995
// MI455X (gfx1250) — compile-verified
//
#include <hip/hip_runtime.h>

// StructuredSparsityMask (gumbel top-2-of-4) — streaming elementwise kernel.
//
// Reference collapses to (tau == 1, straight-through => forward == y_hard):
//   noisy[i] = logits[i] + gumbel(u1[i])
//   hard[i]  = (rank of noisy[i] among 4, ties -> lower index) < 2
//   mask[i]  = (hard[i] + gumbel(u2[i,1]) > gumbel(u2[i,0])) ? 1 : 0
//
// Roofline: 80 B per 4-output block, 671 MB single-use traffic -> ~29 us floor
// at 23.3 TB/s. Round-0 codegen spent ~328 VALU/thread (~34 us) because each
// __logf carried an ocml denorm guard. Here log args are provably normal
// (u+eps in [1e-8,1), inner in (1e-8,18.5]), so we use raw v_log_f32
// (__builtin_amdgcn_logf = log2) + fma constant folding: ~5 VALU per gumbel,
// keeping compute well under the HBM roof. Memory path: 128-bit non-temporal
// loads/stores so the one-pass stream doesn't thrash the 192 MB L2.

typedef __attribute__((ext_vector_type(4))) float v4f;

#define EPS_G 1e-8f
#define LN2_F 0.69314718055994530942f

__device__ __forceinline__ float gumbel(float u) {
    // gumbel(u) = -ln(-ln(u+eps)+eps)
    //           = -ln2 * log2( fma(log2(u+eps), -ln2, eps) )
    const float a     = __builtin_amdgcn_logf(u + EPS_G);   // v_log_f32: log2(u+eps)
    const float inner = __builtin_fmaf(a, -LN2_F, EPS_G);   // -ln(u+eps)+eps  (always > 0, normal)
    const float b     = __builtin_amdgcn_logf(inner);       // v_log_f32: log2(inner)
    return -LN2_F * b;
}

__global__ __launch_bounds__(256) void StructuredSparsityMask_kernel(
    const v4f* __restrict__ logits,   // [NBLK] x 4 f32
    const v4f* __restrict__ u1,       // [NBLK] x 4 f32
    const v4f* __restrict__ u2,       // [NBLK] x 8 f32 (two v4f per block)
    v4f* __restrict__ out,            // [NBLK] x 4 f32
    int nblocks)
{
    const int t = blockIdx.x * blockDim.x + threadIdx.x;
    if (t >= nblocks) return;

    // 128-bit non-temporal streaming loads (global_load_b128, TH=NT)
    const v4f L  = __builtin_nontemporal_load(&logits[t]);
    const v4f U1 = __builtin_nontemporal_load(&u1[t]);
    const v4f A  = __builtin_nontemporal_load(&u2[2 * t + 0]); // i=0,1: (c0,c1,c0,c1)
    const v4f B  = __builtin_nontemporal_load(&u2[2 * t + 1]); // i=2,3: (c0,c1,c0,c1)

    // noisy = logits + gumbel(u1)
    float n[4];
#pragma unroll
    for (int i = 0; i < 4; ++i) n[i] = L[i] + gumbel(U1[i]);

    // top-2 of 4 with lax.top_k tie-breaking (equal values: lower index wins)
    float hard[4];
#pragma unroll
    for (int i = 0; i < 4; ++i) {
        int rank = 0;
#pragma unroll
        for (int j = 0; j < 4; ++j) {
            if (j == i) continue;
            const bool beats = (n[j] > n[i]) || ((n[j] == n[i]) && (j < i));
            rank += beats ? 1 : 0;
        }
        hard[i] = (rank < 2) ? 1.0f : 0.0f;
    }

    // second gumbel-softmax (hard=True) forward value == argmax one-hot:
    //   mask[i] = 1 iff hard[i] + gumbel(c1_i) > gumbel(c0_i)   (strict >)
    const float c0[4] = { A[0], A[2], B[0], B[2] };
    const float c1[4] = { A[1], A[3], B[1], B[3] };

    v4f r;
#pragma unroll
    for (int i = 0; i < 4; ++i) {
        r[i] = ((hard[i] + gumbel(c1[i])) > gumbel(c0[i])) ? 1.0f : 0.0f;
    }

    // 128-bit non-temporal streaming store (global_store_b128, TH=NT)
    __builtin_nontemporal_store(r, &out[t]);
}

extern "C" void kernel_launch(void* const* d_in, const int* in_sizes, int n_in,
                              void* d_out, int out_size, void* d_ws, size_t ws_size,
                              hipStream_t stream) {
    (void)n_in; (void)out_size; (void)d_ws; (void)ws_size;

    const v4f* logits = (const v4f*)d_in[0];  // mask_logits [NBLK,4] f32
    const v4f* u1     = (const v4f*)d_in[1];  // u1          [NBLK,4] f32
    const v4f* u2     = (const v4f*)d_in[2];  // u2          [NBLK,4,2] f32
    v4f*       out    = (v4f*)d_out;          // mask        [4096,8192] f32

    const int nblocks = in_sizes[0] / 4;      // 8,388,608
    const int threads = 256;                  // 8 wave32s per workgroup
    const int grid    = (nblocks + threads - 1) / threads;

    StructuredSparsityMask_kernel<<<grid, threads, 0, stream>>>(logits, u1, u2, out, nblocks);
}